// GraphSAGEEncoder_24257975288372
// MI455X (gfx1250) — compile-verified
//
#include <hip/hip_runtime.h>

typedef __attribute__((ext_vector_type(2))) float v2f;
typedef __attribute__((ext_vector_type(4))) float v4f;
typedef __attribute__((ext_vector_type(8))) float v8f;

#define IN_C  128
#define HID_C 256
#define OUT_C 128

// ---------------------------------------------------------------- zero scratch
__global__ void zero_kernel(float* __restrict__ p, long n4) {
    long i = (long)blockIdx.x * blockDim.x + threadIdx.x;
    long stride = (long)gridDim.x * blockDim.x;
    v4f z = {0.f, 0.f, 0.f, 0.f};
    for (; i < n4; i += stride) ((v4f*)p)[i] = z;
}

// ------------------------------------------------- transpose+pair-pack weights
// Wl, Wr: [Nd][Kh] row-major.  Logical B is [2*Kh][Nd] = [Wl^T ; Wr^T].
// Packed layout: Bp[(k>>1)*(Nd*2) + n*2 + (k&1)]  -> per-lane float2 {B[k][n],B[k+1][n]}
__global__ void pack_b_kernel(const float* __restrict__ Wl, const float* __restrict__ Wr,
                              float* __restrict__ Bp, int Kh, int Nd) {
    int idx = blockIdx.x * blockDim.x + threadIdx.x;
    int total = 2 * Kh * Nd;
    if (idx >= total) return;
    int k = idx / Nd, n = idx % Nd;
    float v = (k < Kh) ? Wl[n * Kh + k] : Wr[n * Kh + (k - Kh)];
    Bp[(k >> 1) * (Nd * 2) + n * 2 + (k & 1)] = v;
}

// ---------------------------------------------------------------- degree count
__global__ void deg_kernel(const int* __restrict__ dst, float* __restrict__ deg, int E) {
    int e = blockIdx.x * blockDim.x + threadIdx.x;
    if (e < E) unsafeAtomicAdd(&deg[dst[e]], 1.0f);
}

__global__ void invdeg_kernel(float* __restrict__ deg, int n) {
    int i = blockIdx.x * blockDim.x + threadIdx.x;
    if (i < n) deg[i] = 1.0f / fmaxf(deg[i], 1.0f);
}

// ------------------------------------------------- edge scatter: sum[dst]+=feat[src]
template <int F>
__global__ void scatter_kernel(const float* __restrict__ feat, const int* __restrict__ src,
                               const int* __restrict__ dst, float* __restrict__ sum, int E) {
    const int TPE = F / 4;
    long tid = (long)blockIdx.x * blockDim.x + threadIdx.x;
    long total = (long)E * TPE;
    if (tid >= total) return;
    int e = (int)(tid / TPE);
    int j = (int)(tid % TPE) * 4;
    int s = src[e], d = dst[e];
    v4f v = *(const v4f*)(feat + (long)s * F + j);
    float* o = sum + (long)d * F + j;
    unsafeAtomicAdd(o + 0, v.x);
    unsafeAtomicAdd(o + 1, v.y);
    unsafeAtomicAdd(o + 2, v.z);
    unsafeAtomicAdd(o + 3, v.w);
}

// ---------------------------------------------------------------- fused SAGE GEMM
// out[M][ND] = act( (sumA*invdeg) @ B[0:KH] + selfA @ B[KH:2KH] + bias )
// One wave32 computes a 16x64 strip with 4x V_WMMA_F32_16X16X4_F32 tiles.
template <int KH, int ND, bool RELU>
__global__ __launch_bounds__(256) void sage_gemm_kernel(
    const float* __restrict__ sumA, const float* __restrict__ invdeg,
    const float* __restrict__ selfA, const float* __restrict__ Bp,
    const float* __restrict__ bias, float* __restrict__ out, int M) {
    const int CG = ND / 64;
    int wave = (int)((blockIdx.x * blockDim.x + threadIdx.x) >> 5);
    int lane = threadIdx.x & 31;
    int rowTile = wave / CG;
    int m0 = rowTile * 16;
    if (m0 >= M) return;
    int n0 = (wave % CG) * 64;

    int n    = lane & 15;        // row (A) / col (B,C) within tile
    int hi   = lane >> 4;        // half-wave select
    int laneK = hi * 2;          // K offset for A/B fragments (16x16x4 layout)
    int row  = m0 + n;

    v8f a0 = {0,0,0,0,0,0,0,0}, a1 = {0,0,0,0,0,0,0,0};
    v8f a2 = {0,0,0,0,0,0,0,0}, a3 = {0,0,0,0,0,0,0,0};

    // ---- phase 1: mean-aggregated neighbors (scale fused into A load)
    {
        float sc = invdeg[row];
        const float* arow = sumA + (long)row * KH;
#pragma unroll 4
        for (int k = 0; k < KH; k += 4) {
            int kk = k + laneK;
            v2f a = *(const v2f*)(arow + kk);
            a.x *= sc; a.y *= sc;
            const float* bp = Bp + (long)(kk >> 1) * (ND * 2);
            v2f b0 = *(const v2f*)(bp + (n0 +  0 + n) * 2);
            v2f b1 = *(const v2f*)(bp + (n0 + 16 + n) * 2);
            v2f b2 = *(const v2f*)(bp + (n0 + 32 + n) * 2);
            v2f b3 = *(const v2f*)(bp + (n0 + 48 + n) * 2);
            a0 = __builtin_amdgcn_wmma_f32_16x16x4_f32(false, a, false, b0, (short)0, a0, false, false);
            a1 = __builtin_amdgcn_wmma_f32_16x16x4_f32(false, a, false, b1, (short)0, a1, false, false);
            a2 = __builtin_amdgcn_wmma_f32_16x16x4_f32(false, a, false, b2, (short)0, a2, false, false);
            a3 = __builtin_amdgcn_wmma_f32_16x16x4_f32(false, a, false, b3, (short)0, a3, false, false);
        }
    }
    // ---- phase 2: self term (B rows offset by KH)
    {
        const float* arow = selfA + (long)row * KH;
#pragma unroll 4
        for (int k = 0; k < KH; k += 4) {
            int kk = k + laneK;
            v2f a = *(const v2f*)(arow + kk);
            const float* bp = Bp + (long)((KH + kk) >> 1) * (ND * 2);
            v2f b0 = *(const v2f*)(bp + (n0 +  0 + n) * 2);
            v2f b1 = *(const v2f*)(bp + (n0 + 16 + n) * 2);
            v2f b2 = *(const v2f*)(bp + (n0 + 32 + n) * 2);
            v2f b3 = *(const v2f*)(bp + (n0 + 48 + n) * 2);
            a0 = __builtin_amdgcn_wmma_f32_16x16x4_f32(false, a, false, b0, (short)0, a0, false, false);
            a1 = __builtin_amdgcn_wmma_f32_16x16x4_f32(false, a, false, b1, (short)0, a1, false, false);
            a2 = __builtin_amdgcn_wmma_f32_16x16x4_f32(false, a, false, b2, (short)0, a2, false, false);
            a3 = __builtin_amdgcn_wmma_f32_16x16x4_f32(false, a, false, b3, (short)0, a3, false, false);
        }
    }
    // ---- epilogue: bias (+ReLU) and store; C layout: VGPR i -> row m0+i(+8 for hi half)
    float bb0 = bias[n0 +  0 + n];
    float bb1 = bias[n0 + 16 + n];
    float bb2 = bias[n0 + 32 + n];
    float bb3 = bias[n0 + 48 + n];
    int rbase = m0 + hi * 8;
#pragma unroll
    for (int i = 0; i < 8; i++) {
        long ro = (long)(rbase + i) * ND + n0 + n;
        float t0 = a0[i] + bb0, t1 = a1[i] + bb1, t2 = a2[i] + bb2, t3 = a3[i] + bb3;
        if (RELU) {
            t0 = fmaxf(t0, 0.f); t1 = fmaxf(t1, 0.f);
            t2 = fmaxf(t2, 0.f); t3 = fmaxf(t3, 0.f);
        }
        out[ro +  0] = t0;
        out[ro + 16] = t1;
        out[ro + 32] = t2;
        out[ro + 48] = t3;
    }
}

extern "C" void kernel_launch(void* const* d_in, const int* in_sizes, int n_in,
                              void* d_out, int out_size, void* d_ws, size_t ws_size,
                              hipStream_t stream) {
    const float* x   = (const float*)d_in[0];
    const int*   ei  = (const int*)d_in[1];
    const float* W1l = (const float*)d_in[2];
    const float* b1  = (const float*)d_in[3];
    const float* W1r = (const float*)d_in[4];
    const float* W2l = (const float*)d_in[5];
    const float* b2  = (const float*)d_in[6];
    const float* W2r = (const float*)d_in[7];
    float* out = (float*)d_out;

    const int N = in_sizes[0] / IN_C;   // 50000
    const int E = in_sizes[1] / 2;      // 800000
    const int* src = ei;
    const int* dst = ei + E;

    // workspace layout (floats)
    float* ws   = (float*)d_ws;
    float* deg  = ws;                                  // N        (becomes invdeg)
    float* sum1 = deg  + N;                            // N*IN_C
    float* sum2 = sum1 + (long)N * IN_C;               // N*HID_C
    float* B1p  = sum2 + (long)N * HID_C;              // 2*IN_C*HID_C
    float* B2p  = B1p  + 2 * IN_C * HID_C;             // 2*HID_C*OUT_C
    float* h    = B2p  + 2 * HID_C * OUT_C;            // N*HID_C

    // zero the atomic accumulators (deg | sum1 | sum2 are contiguous)
    long zn4 = (long)N * (1 + IN_C + HID_C) / 4;
    zero_kernel<<<2048, 256, 0, stream>>>(ws, zn4);

    // pack weights: B1 = [W1l^T ; W1r^T] (K=256,N=256), B2 = [W2l^T ; W2r^T] (K=512,N=128)
    pack_b_kernel<<<(2 * IN_C * HID_C + 255) / 256, 256, 0, stream>>>(W1l, W1r, B1p, IN_C, HID_C);
    pack_b_kernel<<<(2 * HID_C * OUT_C + 255) / 256, 256, 0, stream>>>(W2l, W2r, B2p, HID_C, OUT_C);

    // degrees -> reciprocal
    deg_kernel<<<(E + 255) / 256, 256, 0, stream>>>(dst, deg, E);
    invdeg_kernel<<<(N + 255) / 256, 256, 0, stream>>>(deg, N);

    // layer 1: scatter x, fused GEMM (relu)
    long s1t = (long)E * (IN_C / 4);
    scatter_kernel<IN_C><<<(int)((s1t + 255) / 256), 256, 0, stream>>>(x, src, dst, sum1, E);
    int rt = (N + 15) / 16;
    int waves1 = rt * (HID_C / 64);
    sage_gemm_kernel<IN_C, HID_C, true>
        <<<(waves1 * 32 + 255) / 256, 256, 0, stream>>>(sum1, deg, x, B1p, b1, h, N);

    // layer 2: scatter h, fused GEMM (no relu) -> d_out
    long s2t = (long)E * (HID_C / 4);
    scatter_kernel<HID_C><<<(int)((s2t + 255) / 256), 256, 0, stream>>>(h, src, dst, sum2, E);
    int waves2 = rt * (OUT_C / 64);
    sage_gemm_kernel<HID_C, OUT_C, false>
        <<<(waves2 * 32 + 255) / 256, 256, 0, stream>>>(sum2, deg, h, B2p, b2, out, N);
}